// GraphAttentionLayer_43138651521074
// MI455X (gfx1250) — compile-verified
//
#include <hip/hip_runtime.h>
#include <math.h>

#define N_NODES 100000
#define N_EDGES 1600000
#define F_IN    128
#define F_E     32
#define DH      128   // D*H
#define D_HEAD  32
#define H_HEADS 4

typedef float v2f __attribute__((ext_vector_type(2)));
typedef float v8f __attribute__((ext_vector_type(8)));

// ---------------------------------------------------------------------------
// Init: agg = 0, mx = -inf, denom = 0
// ---------------------------------------------------------------------------
__global__ void k_init(float* __restrict__ agg, float* __restrict__ mx,
                       float* __restrict__ denom) {
    int i = blockIdx.x * blockDim.x + threadIdx.x;
    if (i < N_NODES * DH) agg[i] = 0.0f;
    if (i < N_NODES * H_HEADS) { mx[i] = -__builtin_inff(); denom[i] = 0.0f; }
}

// ---------------------------------------------------------------------------
// Collapse W_edge (32x128) with a_edge (4x32) into W_eh (32x4):
//   W_eh[f][h] = sum_d W_edge[f, h*32+d] * attn_kernel[h, 64+d]
// ---------------------------------------------------------------------------
__global__ void k_weh(const float* __restrict__ W_edge,
                      const float* __restrict__ attn_kernel,
                      float* __restrict__ W_eh) {
    int t = threadIdx.x;          // 128 threads: (f, h)
    if (t >= F_E * H_HEADS) return;
    int f = t >> 2, h = t & 3;
    float s = 0.0f;
    for (int d = 0; d < D_HEAD; ++d)
        s += W_edge[f * DH + h * D_HEAD + d] *
             attn_kernel[h * (3 * D_HEAD) + 2 * D_HEAD + d];
    W_eh[f * H_HEADS + h] = s;
}

// ---------------------------------------------------------------------------
// node_proj = node_features @ W_node   (N x 128) = (N x 128)(128 x 128)
// fp32 WMMA 16x16x4. 256 threads = 8 waves; each wave owns 2 M-tiles
// (32 rows) so every B fragment feeds two WMMAs. W_node staged in 64KB LDS.
// A-fragment loads are UNCONDITIONAL b64 from a row-clamped pointer;
// invalid rows are zeroed with a multiply (no EXEC branching in the loop).
// ---------------------------------------------------------------------------
__global__ __launch_bounds__(256) void k_node_proj(
    const float* __restrict__ X, const float* __restrict__ W,
    float* __restrict__ P) {
    __shared__ float Ws[F_IN * DH];                 // 64 KB
    for (int i = threadIdx.x; i < F_IN * DH; i += 256) Ws[i] = W[i];
    __syncthreads();

    const int wave  = threadIdx.x >> 5;
    const int lane  = threadIdx.x & 31;
    const int rlo   = lane & 15;
    const int khalf = (lane >> 4) * 2;              // 0 or 2
    const int row0  = blockIdx.x * 256 + wave * 32; // 2 M-tiles per wave

    const int   r0  = row0 + rlo;
    const int   r1  = row0 + 16 + rlo;
    const float v0f = (r0 < N_NODES) ? 1.0f : 0.0f;
    const float v1f = (r1 < N_NODES) ? 1.0f : 0.0f;
    const float* x0 = X + (size_t)((r0 < N_NODES) ? r0 : 0) * F_IN + khalf;
    const float* x1 = X + (size_t)((r1 < N_NODES) ? r1 : 0) * F_IN + khalf;

    v8f acc0[8], acc1[8];
    for (int ct = 0; ct < 8; ++ct) {
        acc0[ct] = (v8f){0,0,0,0,0,0,0,0};
        acc1[ct] = (v8f){0,0,0,0,0,0,0,0};
    }

    for (int k = 0; k < F_IN; k += 4) {
        v2f a0 = *(const v2f*)(x0 + k);             // always-safe b64 load
        v2f a1 = *(const v2f*)(x1 + k);
        a0.x *= v0f; a0.y *= v0f;
        a1.x *= v1f; a1.y *= v1f;
        #pragma unroll
        for (int ct = 0; ct < 8; ++ct) {
            v2f b;
            const int col = ct * 16 + rlo;
            b.x = Ws[(k + khalf) * DH + col];
            b.y = Ws[(k + khalf + 1) * DH + col];
            acc0[ct] = __builtin_amdgcn_wmma_f32_16x16x4_f32(
                false, a0, false, b, (short)0, acc0[ct], false, false);
            acc1[ct] = __builtin_amdgcn_wmma_f32_16x16x4_f32(
                false, a1, false, b, (short)0, acc1[ct], false, false);
        }
    }

    #pragma unroll
    for (int ct = 0; ct < 8; ++ct) {
        #pragma unroll
        for (int r = 0; r < 8; ++r) {
            const int m0 = row0 + r + (lane >> 4) * 8;
            const int m1 = m0 + 16;
            if (m0 < N_NODES)
                P[(size_t)m0 * DH + ct * 16 + rlo] = acc0[ct][r];
            if (m1 < N_NODES)
                P[(size_t)m1 * DH + ct * 16 + rlo] = acc1[ct][r];
        }
    }
}

// ---------------------------------------------------------------------------
// s_src[n,h] / s_dst[n,h]: per-head dot of node_proj with a_src / a_dst
// (node_proj is L2-resident; this pass is ~free)
// ---------------------------------------------------------------------------
__global__ void k_scores(const float* __restrict__ P,
                         const float* __restrict__ attn_kernel,
                         float* __restrict__ s_src, float* __restrict__ s_dst) {
    int i = blockIdx.x * blockDim.x + threadIdx.x;    // (n, h)
    if (i >= N_NODES * H_HEADS) return;
    int n = i >> 2, h = i & 3;
    const float* p   = P + (size_t)n * DH + h * D_HEAD;
    const float* asr = attn_kernel + h * (3 * D_HEAD);
    const float* ads = asr + D_HEAD;
    float ss = 0.0f, sd = 0.0f;
    #pragma unroll
    for (int d = 0; d < D_HEAD; ++d) {
        float v = p[d];
        ss += v * asr[d];
        sd += v * ads[d];
    }
    s_src[i] = ss; s_dst[i] = sd;
}

// ---------------------------------------------------------------------------
// Per edge: logits[e,h] = s_src[src,h] + s_dst[dst,h] + ef[e]·W_eh[:,h]
// then float atomicMax into mx[dst,h]  (GLOBAL_ATOMIC_MAX_NUM_F32)
// ---------------------------------------------------------------------------
__global__ __launch_bounds__(256) void k_logits(
    const float* __restrict__ EF, const int* __restrict__ EI,
    const float* __restrict__ W_eh, const float* __restrict__ s_src,
    const float* __restrict__ s_dst, float* __restrict__ logits,
    float* __restrict__ mx) {
    __shared__ float Wl[F_E * H_HEADS];
    if (threadIdx.x < F_E * H_HEADS) Wl[threadIdx.x] = W_eh[threadIdx.x];
    __syncthreads();

    int e = blockIdx.x * blockDim.x + threadIdx.x;
    if (e >= N_EDGES) return;
    const int src = EI[2 * e], dst = EI[2 * e + 1];

    float sh0 = 0.0f, sh1 = 0.0f, sh2 = 0.0f, sh3 = 0.0f;
    const float* ef = EF + (size_t)e * F_E;
    #pragma unroll
    for (int f = 0; f < F_E; ++f) {
        float v = ef[f];
        sh0 += v * Wl[f * 4 + 0];
        sh1 += v * Wl[f * 4 + 1];
        sh2 += v * Wl[f * 4 + 2];
        sh3 += v * Wl[f * 4 + 3];
    }
    float4 l;
    l.x = sh0 + s_src[src * 4 + 0] + s_dst[dst * 4 + 0];
    l.y = sh1 + s_src[src * 4 + 1] + s_dst[dst * 4 + 1];
    l.z = sh2 + s_src[src * 4 + 2] + s_dst[dst * 4 + 2];
    l.w = sh3 + s_src[src * 4 + 3] + s_dst[dst * 4 + 3];
    ((float4*)logits)[e] = l;

    atomicMax(&mx[dst * 4 + 0], l.x);
    atomicMax(&mx[dst * 4 + 1], l.y);
    atomicMax(&mx[dst * 4 + 2], l.z);
    atomicMax(&mx[dst * 4 + 3], l.w);
}

// ---------------------------------------------------------------------------
// Fused exp + aggregation: one wave32 per edge.
//   ex = exp(logit - mx[dst]); one lane per head does atomicAdd(denom);
//   all lanes scatter UNNORMALIZED ex * node_proj[src] into agg[dst]
// (division by denom is folded into k_out, so the denom pass disappears).
// node_proj / agg (51.2 MB each) are L2-resident on MI455X (192 MB L2).
// ---------------------------------------------------------------------------
__global__ __launch_bounds__(256) void k_agg(
    const int* __restrict__ EI, const float* __restrict__ logits,
    const float* __restrict__ mx, float* __restrict__ denom,
    const float* __restrict__ P, float* __restrict__ agg) {
    const int gwave = (blockIdx.x * blockDim.x + threadIdx.x) >> 5;
    const int lane  = threadIdx.x & 31;
    if (gwave >= N_EDGES) return;
    const int e   = gwave;
    const int src = EI[2 * e], dst = EI[2 * e + 1];
    const int h   = lane >> 3;                     // (lane*4)/32

    const float ex = expf(logits[e * 4 + h] - mx[dst * 4 + h]);
    if ((lane & 7) == 0) atomicAdd(&denom[dst * 4 + h], ex);

    const float4 np = ((const float4*)(P + (size_t)src * DH))[lane];
    float* dp = agg + (size_t)dst * DH + lane * 4;
    atomicAdd(dp + 0, ex * np.x);
    atomicAdd(dp + 1, ex * np.y);
    atomicAdd(dp + 2, ex * np.z);
    atomicAdd(dp + 3, ex * np.w);
}

// ---------------------------------------------------------------------------
// out = gelu( (agg / (denom+1e-8)) @ W_out + b_out )   (N x 32)
// fp32 WMMA; the per-(row,head) 1/denom scale (zero for invalid rows) is
// folded into the A fragments. Unconditional b64 A loads from clamped rows.
// 2 M-tiles per wave, exact GELU via erff.
// ---------------------------------------------------------------------------
__global__ __launch_bounds__(256) void k_out(
    const float* __restrict__ A, const float* __restrict__ denom,
    const float* __restrict__ W, const float* __restrict__ bias,
    float* __restrict__ Out) {
    __shared__ float Ws[F_IN * D_HEAD];             // 16 KB
    __shared__ float bs[D_HEAD];
    for (int i = threadIdx.x; i < F_IN * D_HEAD; i += 256) Ws[i] = W[i];
    if (threadIdx.x < D_HEAD) bs[threadIdx.x] = bias[threadIdx.x];
    __syncthreads();

    const int wave  = threadIdx.x >> 5;
    const int lane  = threadIdx.x & 31;
    const int rlo   = lane & 15;
    const int khalf = (lane >> 4) * 2;
    const int row0  = blockIdx.x * 256 + wave * 32; // 2 M-tiles per wave

    const int  r0 = row0 + rlo;
    const int  r1 = row0 + 16 + rlo;
    const bool v0 = (r0 < N_NODES);
    const bool v1 = (r1 < N_NODES);
    const int  c0 = v0 ? r0 : 0;
    const int  c1 = v1 ? r1 : 0;
    const float* a0p = A + (size_t)c0 * F_IN + khalf;
    const float* a1p = A + (size_t)c1 * F_IN + khalf;

    // per-row, per-head reciprocal of (denom + 1e-8); zero for invalid rows
    const float4 d0 = ((const float4*)denom)[c0];
    const float4 d1 = ((const float4*)denom)[c1];
    float s0[4], s1[4];
    s0[0] = v0 ? 1.0f / (d0.x + 1e-8f) : 0.0f;
    s0[1] = v0 ? 1.0f / (d0.y + 1e-8f) : 0.0f;
    s0[2] = v0 ? 1.0f / (d0.z + 1e-8f) : 0.0f;
    s0[3] = v0 ? 1.0f / (d0.w + 1e-8f) : 0.0f;
    s1[0] = v1 ? 1.0f / (d1.x + 1e-8f) : 0.0f;
    s1[1] = v1 ? 1.0f / (d1.y + 1e-8f) : 0.0f;
    s1[2] = v1 ? 1.0f / (d1.z + 1e-8f) : 0.0f;
    s1[3] = v1 ? 1.0f / (d1.w + 1e-8f) : 0.0f;

    v8f acc0[2], acc1[2];
    acc0[0] = (v8f){0,0,0,0,0,0,0,0}; acc0[1] = (v8f){0,0,0,0,0,0,0,0};
    acc1[0] = (v8f){0,0,0,0,0,0,0,0}; acc1[1] = (v8f){0,0,0,0,0,0,0,0};

    #pragma unroll
    for (int h = 0; h < H_HEADS; ++h) {
        const float sc0 = s0[h];
        const float sc1 = s1[h];
        for (int kk = 0; kk < D_HEAD; kk += 4) {
            const int k = h * D_HEAD + kk;
            v2f a0 = *(const v2f*)(a0p + k);        // always-safe b64 load
            v2f a1 = *(const v2f*)(a1p + k);
            a0.x *= sc0; a0.y *= sc0;
            a1.x *= sc1; a1.y *= sc1;
            #pragma unroll
            for (int ct = 0; ct < 2; ++ct) {
                v2f b;
                const int col = ct * 16 + rlo;
                b.x = Ws[(k + khalf) * D_HEAD + col];
                b.y = Ws[(k + khalf + 1) * D_HEAD + col];
                acc0[ct] = __builtin_amdgcn_wmma_f32_16x16x4_f32(
                    false, a0, false, b, (short)0, acc0[ct], false, false);
                acc1[ct] = __builtin_amdgcn_wmma_f32_16x16x4_f32(
                    false, a1, false, b, (short)0, acc1[ct], false, false);
            }
        }
    }

    #pragma unroll
    for (int ct = 0; ct < 2; ++ct) {
        const int col = ct * 16 + rlo;
        #pragma unroll
        for (int r = 0; r < 8; ++r) {
            const int m0 = row0 + r + (lane >> 4) * 8;
            const int m1 = m0 + 16;
            if (m0 < N_NODES) {
                float x = acc0[ct][r] + bs[col];
                Out[(size_t)m0 * D_HEAD + col] =
                    0.5f * x * (1.0f + erff(x * 0.70710678118654752f));
            }
            if (m1 < N_NODES) {
                float x = acc1[ct][r] + bs[col];
                Out[(size_t)m1 * D_HEAD + col] =
                    0.5f * x * (1.0f + erff(x * 0.70710678118654752f));
            }
        }
    }
}

// ---------------------------------------------------------------------------
extern "C" void kernel_launch(void* const* d_in, const int* in_sizes, int n_in,
                              void* d_out, int out_size, void* d_ws, size_t ws_size,
                              hipStream_t stream) {
    const float* node_features = (const float*)d_in[0];
    const int*   edge_index    = (const int*)d_in[1];
    const float* edge_features = (const float*)d_in[2];
    const float* W_node        = (const float*)d_in[3];
    const float* W_edge        = (const float*)d_in[4];
    const float* attn_kernel   = (const float*)d_in[5];
    const float* W_out         = (const float*)d_in[6];
    const float* b_out         = (const float*)d_in[7];
    float* out = (float*)d_out;

    // workspace layout (floats)
    float* ws        = (float*)d_ws;
    float* node_proj = ws;                                   // N*128
    float* s_src     = node_proj + (size_t)N_NODES * DH;     // N*4
    float* s_dst     = s_src + (size_t)N_NODES * H_HEADS;    // N*4
    float* mx        = s_dst + (size_t)N_NODES * H_HEADS;    // N*4
    float* denom     = mx + (size_t)N_NODES * H_HEADS;       // N*4
    float* logits    = denom + (size_t)N_NODES * H_HEADS;    // E*4
    float* agg       = logits + (size_t)N_EDGES * H_HEADS;   // N*128
    float* W_eh      = agg + (size_t)N_NODES * DH;           // 128

    const int initN = N_NODES * DH;                          // covers all init ranges
    k_init<<<(initN + 255) / 256, 256, 0, stream>>>(agg, mx, denom);
    k_weh<<<1, 128, 0, stream>>>(W_edge, attn_kernel, W_eh);

    const int gemmBlocks = (N_NODES + 255) / 256;            // 256 rows per block
    k_node_proj<<<gemmBlocks, 256, 0, stream>>>(node_features, W_node, node_proj);

    k_scores<<<(N_NODES * H_HEADS + 255) / 256, 256, 0, stream>>>(
        node_proj, attn_kernel, s_src, s_dst);

    k_logits<<<(N_EDGES + 255) / 256, 256, 0, stream>>>(
        edge_features, edge_index, W_eh, s_src, s_dst, logits, mx);

    // one wave (32 threads) per edge -> E*32 threads (fused exp + aggregate)
    const long long aggThreads = (long long)N_EDGES * 32;
    k_agg<<<(int)((aggThreads + 255) / 256), 256, 0, stream>>>(
        edge_index, logits, mx, denom, node_proj, agg);

    k_out<<<gemmBlocks, 256, 0, stream>>>(agg, denom, W_out, b_out, out);
}